// Region_80479097193021
// MI455X (gfx1250) — compile-verified
//
#include <hip/hip_runtime.h>
#include <hip/hip_bf16.h>
#include <math.h>

// ---------------------------------------------------------------------------
// Sizes (from reference)
// ---------------------------------------------------------------------------
#define N_NODES 10000
#define MEMD    128
#define LAMB_LEN 2
#define HEADS   8
#define FPH     16
#define N_REG   100
#define N_GRAPH 4
#define NODE_FEAT 64
#define MSG_DIM 128
#define RAWD    320         // MEM*LAMB_LEN + NODE_FEAT
#define AGG_C   321         // RAWD + 1 (ones column)
#define BEDGE   100000
#define LAMB0   1.0f
#define LAMB1   0.1f
#define KCHUNK  16          // K rows of W staged to LDS per async chunk

typedef float v2f __attribute__((ext_vector_type(2)));
typedef float v8f __attribute__((ext_vector_type(8)));

__device__ __forceinline__ float lrelu_f(float x) { return x > 0.f ? x : 0.01f * x; }

// ---------------------------------------------------------------------------
// Zero a float buffer
// ---------------------------------------------------------------------------
__global__ void zero_kernel(float* p, size_t count) {
    size_t i = (size_t)blockIdx.x * blockDim.x + threadIdx.x;
    if (i < count) p[i] = 0.f;
}

// ---------------------------------------------------------------------------
// Edge scatter: agg[src, l, j] += wts[l] * raw_msg[j]   (atomics stay in L2)
// ---------------------------------------------------------------------------
__global__ void edge_scatter(const int* __restrict__ src, const int* __restrict__ tgt,
                             const float* __restrict__ etd, const float* __restrict__ ts,
                             const float* __restrict__ now_t,
                             const float* __restrict__ memory,
                             const float* __restrict__ node_features,
                             float* __restrict__ agg) {
    int e = blockIdx.x;
    int t = threadIdx.x;
    int s = src[e], d = tgt[e];
    float now = now_t[0];
    float td  = etd[e];
    float dec0 = expf(-LAMB0 * td), dec1 = expf(-LAMB1 * td);
    float dt   = now - ts[e];
    float w0 = expf(-LAMB0 * dt), w1 = expf(-LAMB1 * dt);
    float* aggs = agg + (size_t)s * (2 * AGG_C);
    for (int j = t; j < AGG_C; j += blockDim.x) {
        float val;
        if (j < 128)      val = memory[((size_t)d * 2 + 0) * 128 + j] * dec0;
        else if (j < 256) val = memory[((size_t)d * 2 + 1) * 128 + (j - 128)] * dec1;
        else if (j < 320) val = node_features[(size_t)d * 64 + (j - 256)];
        else              val = 1.0f;
        atomicAdd(&aggs[j], w0 * val);
        atomicAdd(&aggs[AGG_C + j], w1 * val);
    }
}

// ---------------------------------------------------------------------------
// Per-node prep: avg_msg, static_emb (attention mix of static0 and features)
// one block (128 threads) per node
// ---------------------------------------------------------------------------
__global__ void node_prep(const float* __restrict__ agg, const float* __restrict__ memory,
                          const float* __restrict__ last_update,
                          const float* __restrict__ features,
                          const float* __restrict__ fut1_w, const float* __restrict__ fut1_b,
                          const float* __restrict__ fut2_w, const float* __restrict__ fut2_b,
                          const float* __restrict__ begin_t,
                          float* __restrict__ avgmsg, float* __restrict__ static_emb) {
    int n = blockIdx.x, t = threadIdx.x;
    __shared__ float red[128];
    __shared__ float a0s, a1s;
    float begin = begin_t[0];
    float lu = last_update[n];
    float d0 = expf(-LAMB0 * (begin - lu));
    float d1 = expf(-LAMB1 * (begin - lu));
    float sa = memory[(size_t)n * 256 + t] * d0;
    float sb = memory[(size_t)n * 256 + 128 + t] * d1;
    float fa = features[(size_t)n * 256 + t];
    float fb = features[(size_t)n * 256 + 128 + t];
    float p = sa * fut1_w[t] + sb * fut1_w[128 + t];
    red[t] = p; __syncthreads();
    for (int s = 64; s > 0; s >>= 1) { if (t < s) red[t] += red[t + s]; __syncthreads(); }
    if (t == 0) a0s = red[0] + fut1_b[0];
    __syncthreads();
    p = fa * fut2_w[t] + fb * fut2_w[128 + t];
    red[t] = p; __syncthreads();
    for (int s = 64; s > 0; s >>= 1) { if (t < s) red[t] += red[t + s]; __syncthreads(); }
    if (t == 0) a1s = red[0] + fut2_b[0];
    __syncthreads();
    float a0 = a0s, a1 = a1s;
    float m = fmaxf(a0, a1);
    float e0 = expf(a0 - m), e1 = expf(a1 - m);
    float inv = 1.f / (e0 + e1);
    float att0 = e0 * inv, att1 = e1 * inv;
    static_emb[(size_t)n * 256 + t]       = att0 * sa + att1 * fa;
    static_emb[(size_t)n * 256 + 128 + t] = att0 * sb + att1 * fb;
    float ws0 = fmaxf(agg[(size_t)n * 642 + 320], 1e-6f);
    float ws1 = fmaxf(agg[(size_t)n * 642 + AGG_C + 320], 1e-6f);
    for (int idx = t; idx < 640; idx += 128) {
        int l = idx / 320, j = idx % 320;
        float w = l ? ws1 : ws0;
        avgmsg[((size_t)n * 2 + l) * 320 + j] = agg[(size_t)n * 642 + l * AGG_C + j] / w;
    }
}

// ---------------------------------------------------------------------------
// WMMA fp32 GEMM strip kernel: C[M,128] = act(A[M,K] @ W[K,128] + bias).
// One wave (32 lanes) computes a 16x128 strip via 8 accumulators, reusing the
// A fragment across 8 V_WMMA_F32_16X16X4_F32 issues per k-step.
// Weights are staged chunk-wise into LDS with GLOBAL_LOAD_ASYNC_TO_LDS_B128
// (ASYNCcnt) and consumed via ds_load; wave-local counters, so no barriers.
// Requires: M % 16 == 0, K % KCHUNK == 0, W row-major with ldw == 128.
// ---------------------------------------------------------------------------
__global__ void __launch_bounds__(32) wmma_gemm_f32(const float* __restrict__ A, int lda,
                                                    const float* __restrict__ W,
                                                    const float* __restrict__ bias,
                                                    float* __restrict__ C, int ldc,
                                                    int K, int act) {
    __shared__ float wlds[KCHUNK * 128];          // 8 KB staging buffer
    const int m0 = blockIdx.x * 16;
    const int lane = threadIdx.x;                 // 0..31
    const int lr   = lane & 15;
    const int koff = (lane >> 4) * 2;             // lanes 16-31: K+2,K+3 (ISA layout)
    v8f zero = {0.f, 0.f, 0.f, 0.f, 0.f, 0.f, 0.f, 0.f};
    v8f acc[8];
    #pragma unroll
    for (int t = 0; t < 8; ++t) acc[t] = zero;
    const float* arow = A + (size_t)(m0 + lr) * lda;

    for (int kb = 0; kb < K; kb += KCHUNK) {
        // all ds reads of the previous chunk must be done before overwrite
        asm volatile("s_wait_dscnt 0x0" ::: "memory");
        // stage W[kb:kb+KCHUNK, 0:128] (contiguous 2048 floats) into LDS
        const float* wsrc = W + (size_t)kb * 128;
        #pragma unroll
        for (int it = 0; it < (KCHUNK * 128) / (32 * 4); ++it) {  // 16 b128/lane
            int f4 = lane + 32 * it;              // float4 index within chunk
            unsigned ldsb = (unsigned)(size_t)(&wlds[f4 * 4]);
            unsigned long long ga = (unsigned long long)(size_t)(wsrc + f4 * 4);
            asm volatile("global_load_async_to_lds_b128 %0, %1, off"
                         :: "v"(ldsb), "v"(ga) : "memory");
        }
        asm volatile("s_wait_asynccnt 0x0" ::: "memory");
        #pragma unroll
        for (int kk = 0; kk < KCHUNK; kk += 4) {
            v2f a;
            a.x = arow[kb + kk + koff];
            a.y = arow[kb + kk + koff + 1];
            #pragma unroll
            for (int t = 0; t < 8; ++t) {
                v2f b;
                b.x = wlds[(kk + koff) * 128 + t * 16 + lr];
                b.y = wlds[(kk + koff + 1) * 128 + t * 16 + lr];
                acc[t] = __builtin_amdgcn_wmma_f32_16x16x4_f32(false, a, false, b,
                                                               (short)0, acc[t],
                                                               false, false);
            }
        }
    }
    const int rbase = m0 + (lane >> 4) * 8;
    #pragma unroll
    for (int t = 0; t < 8; ++t) {
        int col = t * 16 + lr;
        float bv = bias ? bias[col] : 0.f;
        #pragma unroll
        for (int g = 0; g < 8; ++g) {
            float v = acc[t][g] + bv;
            if (act == 1)      v = fmaxf(v, 0.f);
            else if (act == 2) v = lrelu_f(v);
            C[(size_t)(rbase + g) * ldc + col] = v;
        }
    }
}

// ---------------------------------------------------------------------------
// Small scalar matmul (for N_REG/N_GRAPH-sized ops): one thread per output
// ---------------------------------------------------------------------------
__global__ void small_matmul(const float* __restrict__ A, int lda,
                             const float* __restrict__ W, int ldw,
                             const float* __restrict__ bias,
                             const float* __restrict__ res,
                             float* __restrict__ C, int ldc,
                             int M, int N, int K, int act) {
    int idx = blockIdx.x * blockDim.x + threadIdx.x;
    if (idx >= M * N) return;
    int r = idx / N, c = idx % N;
    float acc = bias ? bias[c] : 0.f;
    for (int k = 0; k < K; ++k) acc += A[(size_t)r * lda + k] * W[(size_t)k * ldw + c];
    if (res) acc += res[(size_t)r * ldc + c];
    if (act == 1)      acc = fmaxf(acc, 0.f);
    else if (act == 2) acc = lrelu_f(acc);
    C[(size_t)r * ldc + c] = acc;
}

// ---------------------------------------------------------------------------
// Region self-attention: remb1 = remb0 + softmax(q k^T/8 + mask) @ v
// ---------------------------------------------------------------------------
__global__ void region_attention(const float* __restrict__ remb0, const float* __restrict__ q,
                                 const float* __restrict__ k, const float* __restrict__ v,
                                 const float* __restrict__ adj, float* __restrict__ out) {
    int i = blockIdx.x, t = threadIdx.x;
    __shared__ float sc[N_REG];
    __shared__ float qs[64];
    if (t < 64) qs[t] = q[i * 64 + t];
    __syncthreads();
    if (t < N_REG) {
        float s = 0.f;
        for (int f = 0; f < 64; ++f) s += qs[f] * k[t * 64 + f];
        s *= 0.125f;
        if (!(adj[i * N_REG + t] > 0.f)) s += -1e9f;
        sc[t] = s;
    }
    __syncthreads();
    if (t == 0) {
        float m = -1e30f;
        for (int j = 0; j < N_REG; ++j) m = fmaxf(m, sc[j]);
        float ssum = 0.f;
        for (int j = 0; j < N_REG; ++j) { sc[j] = expf(sc[j] - m); ssum += sc[j]; }
        float inv = 1.f / ssum;
        for (int j = 0; j < N_REG; ++j) sc[j] *= inv;
    }
    __syncthreads();
    for (int c = t; c < 256; c += 128) {
        float acc = remb0[i * 256 + c];
        for (int j = 0; j < N_REG; ++j) acc += sc[j] * v[j * 256 + c];
        out[i * 256 + c] = acc;
    }
}

// ---------------------------------------------------------------------------
// A_r[(r*8+h), n] = dot(qr[r,h,:16], kr[n,h,:16]) / 4 ; one thread per (n,r)
// ---------------------------------------------------------------------------
__global__ void a_r_kernel(const float* __restrict__ qr, const float* __restrict__ kr,
                           float* __restrict__ A_r) {
    int idx = blockIdx.x * blockDim.x + threadIdx.x;
    if (idx >= N_NODES * N_REG) return;
    int n = idx / N_REG, r = idx % N_REG;
    const float* kn = kr + (size_t)n * 128;
    const float* qq = qr + (size_t)r * 128;
    for (int h = 0; h < HEADS; ++h) {
        float s = 0.f;
        for (int f = 0; f < FPH; ++f) s += qq[h * 16 + f] * kn[h * 16 + f];
        A_r[((size_t)(r * 8 + h)) * N_NODES + n] = s * 0.25f;
    }
}

// meanA[n, r] = mean_h A_r[(r*8+h), n]   (before in-place softmax of A_r)
__global__ void mean_a_kernel(const float* __restrict__ A_r, float* __restrict__ meanA) {
    int idx = blockIdx.x * blockDim.x + threadIdx.x;
    if (idx >= N_NODES * N_REG) return;
    int n = idx / N_REG, r = idx % N_REG;
    float s = 0.f;
    for (int h = 0; h < HEADS; ++h) s += A_r[((size_t)(r * 8 + h)) * N_NODES + n];
    meanA[(size_t)n * N_REG + r] = s * 0.125f;
}

// A_g[(g*8+h), r] = dot(gq[g,h,:16], kg[r,h,:16]) / 4
__global__ void a_g_kernel(const float* __restrict__ gq, const float* __restrict__ kg,
                           float* __restrict__ A_g) {
    int idx = blockIdx.x * blockDim.x + threadIdx.x;
    if (idx >= N_GRAPH * HEADS * N_REG) return;
    int gh = idx / N_REG, r = idx % N_REG;
    int g = gh / 8, h = gh % 8;
    float s = 0.f;
    for (int f = 0; f < FPH; ++f) s += gq[g * 128 + h * 16 + f] * kg[r * 128 + h * 16 + f];
    A_g[(size_t)gh * N_REG + r] = s * 0.25f;
}

// meanAg[g, r] = mean_h A_g[(g*8+h), r]
__global__ void mean_ag_kernel(const float* __restrict__ A_g, float* __restrict__ meanAg) {
    int idx = blockIdx.x * blockDim.x + threadIdx.x;
    if (idx >= N_GRAPH * N_REG) return;
    int g = idx / N_REG, r = idx % N_REG;
    float s = 0.f;
    for (int h = 0; h < HEADS; ++h) s += A_g[(size_t)(g * 8 + h) * N_REG + r];
    meanAg[(size_t)g * N_REG + r] = s * 0.125f;
}

// ---------------------------------------------------------------------------
// Row softmax, in place. One block (256 thr) per row.
// ---------------------------------------------------------------------------
__global__ void softmax_rows(float* __restrict__ X, int ncols) {
    float* row = X + (size_t)blockIdx.x * ncols;
    __shared__ float red[256];
    int t = threadIdx.x;
    float m = -1e30f;
    for (int j = t; j < ncols; j += 256) m = fmaxf(m, row[j]);
    red[t] = m; __syncthreads();
    for (int s = 128; s > 0; s >>= 1) { if (t < s) red[t] = fmaxf(red[t], red[t + s]); __syncthreads(); }
    m = red[0]; __syncthreads();
    float sum = 0.f;
    for (int j = t; j < ncols; j += 256) { float e = expf(row[j] - m); row[j] = e; sum += e; }
    red[t] = sum; __syncthreads();
    for (int s = 128; s > 0; s >>= 1) { if (t < s) red[t] += red[t + s]; __syncthreads(); }
    float inv = 1.f / red[0];
    for (int j = t; j < ncols; j += 256) row[j] *= inv;
}

// ---------------------------------------------------------------------------
// region_mid[r,l,h*16+f] = sum_n P[(r*8+h),n] * Vr[(n*2+l)*128 + h*16+f]
// ---------------------------------------------------------------------------
__global__ void region_mid_kernel(const float* __restrict__ P, const float* __restrict__ Vr,
                                  float* __restrict__ mid) {
    int rh = blockIdx.x;
    int r = rh / 8, h = rh % 8;
    __shared__ float acc[32];
    if (threadIdx.x < 32) acc[threadIdx.x] = 0.f;
    __syncthreads();
    float a[2][16];
    #pragma unroll
    for (int l = 0; l < 2; ++l)
        #pragma unroll
        for (int f = 0; f < 16; ++f) a[l][f] = 0.f;
    const float* Prow = P + (size_t)rh * N_NODES;
    for (int n = threadIdx.x; n < N_NODES; n += 256) {
        float p = Prow[n];
        const float* v0 = Vr + ((size_t)n * 2 + 0) * 128 + h * 16;
        const float* v1 = Vr + ((size_t)n * 2 + 1) * 128 + h * 16;
        #pragma unroll
        for (int f = 0; f < 16; ++f) { a[0][f] += p * v0[f]; a[1][f] += p * v1[f]; }
    }
    #pragma unroll
    for (int l = 0; l < 2; ++l)
        #pragma unroll
        for (int f = 0; f < 16; ++f) atomicAdd(&acc[l * 16 + f], a[l][f]);
    __syncthreads();
    if (threadIdx.x < 32) {
        int l = threadIdx.x / 16, f = threadIdx.x % 16;
        mid[((size_t)r * 2 + l) * 128 + h * 16 + f] = acc[threadIdx.x];
    }
}

// graph_mid[(g*2+l)*128+c] = sum_r Pg[(g*8+h),r] * Vg[(r*2+l)*128+c],  h=c/16
__global__ void graph_mid_kernel(const float* __restrict__ Pg, const float* __restrict__ Vg,
                                 float* __restrict__ gmid) {
    int idx = blockIdx.x * blockDim.x + threadIdx.x;
    if (idx >= N_GRAPH * 2 * 128) return;
    int g = idx / 256;
    int rem = idx % 256;
    int l = rem / 128, c = rem % 128;
    int h = c / 16;
    float s = 0.f;
    for (int r = 0; r < N_REG; ++r)
        s += Pg[(size_t)(g * 8 + h) * N_REG + r] * Vg[((size_t)r * 2 + l) * 128 + c];
    gmid[((size_t)g * 2 + l) * 128 + c] = s;
}

// out = mem * exp(-lamb_l*(now-begin)) + msgs   (region/graph memory update)
__global__ void decay_add(const float* __restrict__ mem, const float* __restrict__ msgs,
                          const float* __restrict__ now_t, const float* __restrict__ begin_t,
                          float* __restrict__ out, int total) {
    int idx = blockIdx.x * blockDim.x + threadIdx.x;
    if (idx >= total) return;
    int l = (idx >> 7) & 1;
    float lamb = l ? LAMB1 : LAMB0;
    float dec = expf(-lamb * (now_t[0] - begin_t[0]));
    out[idx] = mem[idx] * dec + msgs[idx];
}

// updmem_dec[n, l*128+m] = (memory*dec_u + msg*has) * dec_f
__global__ void update_nodes(const float* __restrict__ agg, const float* __restrict__ memory,
                             const float* __restrict__ last_update, const float* __restrict__ msg,
                             const float* __restrict__ now_t, float* __restrict__ updmem_dec) {
    int n = blockIdx.x, t = threadIdx.x;
    float now = now_t[0];
    bool has = agg[(size_t)n * 642 + 320] > 0.f;
    float lu = last_update[n];
    float ul = has ? now : lu;
    #pragma unroll
    for (int l = 0; l < 2; ++l) {
        float lamb = l ? LAMB1 : LAMB0;
        float du = expf(-lamb * (now - lu));
        float df = expf(-lamb * (now - ul));
        float um = memory[(size_t)n * 256 + l * 128 + t] * du +
                   (has ? msg[((size_t)n * 2 + l) * 128 + t] : 0.f);
        updmem_dec[(size_t)n * 256 + l * 128 + t] = um * df;
    }
}

// ge2[r,c] = sum_g softmax_g(meanAg[:,r])[g] * graph_e[g,c]
__global__ void ge2_kernel(const float* __restrict__ meanAg, const float* __restrict__ graph_e,
                           float* __restrict__ ge2) {
    int idx = blockIdx.x * blockDim.x + threadIdx.x;
    if (idx >= N_REG * 128) return;
    int r = idx / 128, c = idx % 128;
    float v[4], m = -1e30f;
    for (int g = 0; g < 4; ++g) { v[g] = meanAg[g * N_REG + r]; m = fmaxf(m, v[g]); }
    float s = 0.f;
    for (int g = 0; g < 4; ++g) { v[g] = expf(v[g] - m); s += v[g]; }
    float inv = 1.f / s, acc = 0.f;
    for (int g = 0; g < 4; ++g) acc += v[g] * inv * graph_e[g * 128 + c];
    ge2[(size_t)r * 128 + c] = acc;
}

// node_region/node_graph into cat columns [128..383]; block per node
__global__ void node_mix(const float* __restrict__ meanA, const float* __restrict__ region_e,
                         const float* __restrict__ ge2, float* __restrict__ cat) {
    int n = blockIdx.x, t = threadIdx.x;
    __shared__ float w[N_REG];
    if (t < N_REG) w[t] = meanA[(size_t)n * N_REG + t];
    __syncthreads();
    if (t == 0) {
        float m = -1e30f;
        for (int r = 0; r < N_REG; ++r) m = fmaxf(m, w[r]);
        float s = 0.f;
        for (int r = 0; r < N_REG; ++r) { w[r] = expf(w[r] - m); s += w[r]; }
        float inv = 1.f / s;
        for (int r = 0; r < N_REG; ++r) w[r] *= inv;
    }
    __syncthreads();
    float nr = 0.f, ng = 0.f;
    for (int r = 0; r < N_REG; ++r) {
        nr += w[r] * region_e[r * 128 + t];
        ng += w[r] * ge2[r * 128 + t];
    }
    cat[(size_t)n * 384 + 128 + t] = nr;
    cat[(size_t)n * 384 + 256 + t] = ng;
}

// ---------------------------------------------------------------------------
// Host launcher
// ---------------------------------------------------------------------------
extern "C" void kernel_launch(void* const* d_in, const int* in_sizes, int n_in,
                              void* d_out, int out_size, void* d_ws, size_t ws_size,
                              hipStream_t stream) {
    const int*   src          = (const int*)d_in[0];
    const int*   tgt          = (const int*)d_in[1];
    const float* etd          = (const float*)d_in[3];
    const float* ts           = (const float*)d_in[4];
    const float* now_t        = (const float*)d_in[5];
    const float* begin_t      = (const float*)d_in[6];
    const float* memory       = (const float*)d_in[7];
    const float* last_update  = (const float*)d_in[8];
    const float* region_mem   = (const float*)d_in[9];
    const float* graph_mem    = (const float*)d_in[10];
    const float* node_feats   = (const float*)d_in[11];
    const float* features     = (const float*)d_in[12];
    const float* adj          = (const float*)d_in[13];
    const float* Wq_r         = (const float*)d_in[14];
    const float* Wk_r         = (const float*)d_in[15];
    const float* Wv_r         = (const float*)d_in[16];
    const float* Wq_g         = (const float*)d_in[17];
    const float* Wk_g         = (const float*)d_in[18];
    const float* Wv_g         = (const float*)d_in[19];
    const float* fut1_w       = (const float*)d_in[20];
    const float* fut1_b       = (const float*)d_in[21];
    const float* fut2_w       = (const float*)d_in[22];
    const float* fut2_b       = (const float*)d_in[23];
    const float* msg_w1       = (const float*)d_in[24];
    const float* msg_b1       = (const float*)d_in[25];
    const float* msg_w2       = (const float*)d_in[26];
    const float* msg_b2       = (const float*)d_in[27];
    const float* ffr_w1       = (const float*)d_in[28];
    const float* ffr_b1       = (const float*)d_in[29];
    const float* ffr_w2       = (const float*)d_in[30];
    const float* ffr_b2       = (const float*)d_in[31];
    const float* ffg_w1       = (const float*)d_in[32];
    const float* ffg_b1       = (const float*)d_in[33];
    const float* ffg_w2       = (const float*)d_in[34];
    const float* ffg_b2       = (const float*)d_in[35];
    const float* embt_w       = (const float*)d_in[36];
    const float* embt_b       = (const float*)d_in[37];
    const float* spat_w       = (const float*)d_in[38];
    const float* spat_b       = (const float*)d_in[39];
    const float* rq_w         = (const float*)d_in[40];
    const float* rk_w         = (const float*)d_in[41];
    const float* rv_w         = (const float*)d_in[42];
    const float* rm_w1        = (const float*)d_in[43];
    const float* rm_b1        = (const float*)d_in[44];
    const float* rm_w2        = (const float*)d_in[45];
    const float* rm_b2        = (const float*)d_in[46];

    float* ws = (float*)d_ws;
    // ---- workspace layout (floats) ----
    float* agg        = ws;                          // 6,420,000
    float* avgmsg     = agg        + 6420000;        // 6,400,000
    float* static_emb = avgmsg     + 6400000;        // 2,560,000
    float* msg1       = static_emb + 2560000;        // 2,560,000
    float* msgb       = msg1       + 2560000;        // 2,560,000
    float* kr         = msgb       + 2560000;        // 1,280,000
    float* Vr         = kr         + 1280000;        // 2,560,000
    float* A_r        = Vr         + 2560000;        // 8,000,000
    float* meanA      = A_r        + 8000000;        // 1,000,000
    float* updmem     = meanA      + 1000000;        // 2,560,000
    float* catb       = updmem     + 2560000;        // 3,840,000
    float* sm         = catb       + 3840000;        // small-region area
    float* qb      = sm;             // 6400
    float* kb      = sm + 6400;      // 6400
    float* vb      = sm + 12800;     // 25600
    float* remb1   = sm + 38400;     // 25600
    float* rhb     = sm + 64000;     // 12800
    float* remb2   = sm + 76800;     // 25600
    float* qr      = sm + 102400;    // 12800
    float* kg      = sm + 115200;    // 12800
    float* gq      = sm + 128000;    // 512
    float* A_g     = sm + 128512;    // 3200
    float* meanAg  = sm + 131712;    // 400
    float* rmid    = sm + 132112;    // 25600
    float* rtmp    = sm + 157712;    // 25600
    float* rmsg    = sm + 183312;    // 25600
    float* Vg      = sm + 208912;    // 25600
    float* gmid    = sm + 234512;    // 1024
    float* gtmp    = sm + 235536;    // 1024
    float* gmsg    = sm + 236560;    // 1024
    float* uR      = sm + 237584;    // 25600
    float* uG      = sm + 263184;    // 1024
    float* reg_e   = sm + 264208;    // 12800
    float* gra_e   = sm + 277008;    // 512
    float* ge2b    = sm + 277520;    // 12800

    float* out = (float*)d_out;

    // 1. zero agg
    {
        size_t cnt = (size_t)N_NODES * 2 * AGG_C;
        zero_kernel<<<(unsigned)((cnt + 255) / 256), 256, 0, stream>>>(agg, cnt);
    }
    // 2. edge scatter
    edge_scatter<<<BEDGE, 128, 0, stream>>>(src, tgt, etd, ts, now_t, memory, node_feats, agg);
    // 3. node prep
    node_prep<<<N_NODES, 128, 0, stream>>>(agg, memory, last_update, features,
                                           fut1_w, fut1_b, fut2_w, fut2_b, begin_t,
                                           avgmsg, static_emb);
    // 4. msg1 = relu(msg_in @ msg_w1 + b1)   [20000 x 320 -> 128]
    wmma_gemm_f32<<<20000 / 16, 32, 0, stream>>>(agg, AGG_C, msg_w1, msg_b1,
                                                 msg1, 128, 320, 1);
    // 5. msg = msg1 @ msg_w2 + b2
    wmma_gemm_f32<<<1250, 32, 0, stream>>>(msg1, 128, msg_w2, msg_b2, msgb, 128, 128, 0);
    // 6. kr = static_emb @ Wk_r
    wmma_gemm_f32<<<625, 32, 0, stream>>>(static_emb, 256, Wk_r, nullptr, kr, 128, 256, 0);
    // 7. Vr = avg_msg @ Wv_r
    wmma_gemm_f32<<<1250, 32, 0, stream>>>(avgmsg, 320, Wv_r, nullptr, Vr, 128, 320, 0);
    // 8-10. region q/k/v
    small_matmul<<<(100 * 64 + 255) / 256, 256, 0, stream>>>(region_mem, 256, rq_w, 64,
        nullptr, nullptr, qb, 64, 100, 64, 256, 0);
    small_matmul<<<(100 * 64 + 255) / 256, 256, 0, stream>>>(region_mem, 256, rk_w, 64,
        nullptr, nullptr, kb, 64, 100, 64, 256, 0);
    small_matmul<<<(100 * 256 + 255) / 256, 256, 0, stream>>>(region_mem, 256, rv_w, 256,
        nullptr, nullptr, vb, 256, 100, 256, 256, 0);
    // 11. region attention
    region_attention<<<N_REG, 128, 0, stream>>>(region_mem, qb, kb, vb, adj, remb1);
    // 12-13. region MLP with residual
    small_matmul<<<(100 * 128 + 255) / 256, 256, 0, stream>>>(remb1, 256, rm_w1, 128,
        rm_b1, nullptr, rhb, 128, 100, 128, 256, 1);
    small_matmul<<<(100 * 256 + 255) / 256, 256, 0, stream>>>(rhb, 128, rm_w2, 256,
        rm_b2, remb1, remb2, 256, 100, 256, 128, 0);
    // 14-16. qr / kg / gq
    small_matmul<<<(100 * 128 + 255) / 256, 256, 0, stream>>>(remb2, 256, Wq_r, 128,
        nullptr, nullptr, qr, 128, 100, 128, 256, 0);
    small_matmul<<<(100 * 128 + 255) / 256, 256, 0, stream>>>(remb2, 256, Wk_g, 128,
        nullptr, nullptr, kg, 128, 100, 128, 256, 0);
    small_matmul<<<(4 * 128 + 255) / 256, 256, 0, stream>>>(graph_mem, 256, Wq_g, 128,
        nullptr, nullptr, gq, 128, 4, 128, 256, 0);
    // 17-19. A_r, meanA, softmax rows of A_r
    a_r_kernel<<<(N_NODES * N_REG + 255) / 256, 256, 0, stream>>>(qr, kr, A_r);
    mean_a_kernel<<<(N_NODES * N_REG + 255) / 256, 256, 0, stream>>>(A_r, meanA);
    softmax_rows<<<800, 256, 0, stream>>>(A_r, N_NODES);
    // 20-22. A_g, meanAg, softmax rows of A_g
    a_g_kernel<<<(N_GRAPH * HEADS * N_REG + 255) / 256, 256, 0, stream>>>(gq, kg, A_g);
    mean_ag_kernel<<<(N_GRAPH * N_REG + 255) / 256, 256, 0, stream>>>(A_g, meanAg);
    softmax_rows<<<32, 256, 0, stream>>>(A_g, N_REG);
    // 23. region_mid
    region_mid_kernel<<<800, 256, 0, stream>>>(A_r, Vr, rmid);
    // 24-25. region message MLP (lrelu twice)
    small_matmul<<<(200 * 128 + 255) / 256, 256, 0, stream>>>(rmid, 128, ffr_w1, 128,
        ffr_b1, nullptr, rtmp, 128, 200, 128, 128, 2);
    small_matmul<<<(200 * 128 + 255) / 256, 256, 0, stream>>>(rtmp, 128, ffr_w2, 128,
        ffr_b2, nullptr, rmsg, 128, 200, 128, 128, 2);
    // 26. Vg = region_mid @ Wv_g
    small_matmul<<<(200 * 128 + 255) / 256, 256, 0, stream>>>(rmid, 128, Wv_g, 128,
        nullptr, nullptr, Vg, 128, 200, 128, 128, 0);
    // 27. graph_mid
    graph_mid_kernel<<<(N_GRAPH * 2 * 128 + 255) / 256, 256, 0, stream>>>(A_g, Vg, gmid);
    // 28-29. graph message MLP
    small_matmul<<<(8 * 128 + 255) / 256, 256, 0, stream>>>(gmid, 128, ffg_w1, 128,
        ffg_b1, nullptr, gtmp, 128, 8, 128, 128, 2);
    small_matmul<<<(8 * 128 + 255) / 256, 256, 0, stream>>>(gtmp, 128, ffg_w2, 128,
        ffg_b2, nullptr, gmsg, 128, 8, 128, 128, 2);
    // 30. region/graph memory updates
    decay_add<<<(25600 + 255) / 256, 256, 0, stream>>>(region_mem, rmsg, now_t, begin_t, uR, 25600);
    decay_add<<<(1024 + 255) / 256, 256, 0, stream>>>(graph_mem, gmsg, now_t, begin_t, uG, 1024);
    // 31-32. region_e / graph_e
    small_matmul<<<(100 * 128 + 255) / 256, 256, 0, stream>>>(uR, 256, embt_w, 128,
        embt_b, nullptr, reg_e, 128, 100, 128, 256, 2);
    small_matmul<<<(4 * 128 + 255) / 256, 256, 0, stream>>>(uG, 256, embt_w, 128,
        embt_b, nullptr, gra_e, 128, 4, 128, 256, 2);
    // 33. ge2 = w_gr^T @ graph_e
    ge2_kernel<<<(N_REG * 128 + 255) / 256, 256, 0, stream>>>(meanAg, gra_e, ge2b);
    // 34. node memory update (decayed)
    update_nodes<<<N_NODES, 128, 0, stream>>>(agg, memory, last_update, msgb, now_t, updmem);
    // 35. node_e -> cat[:, 0:128]
    wmma_gemm_f32<<<625, 32, 0, stream>>>(updmem, 256, embt_w, embt_b, catb, 384, 256, 2);
    // 36. node_region / node_graph -> cat[:, 128:384]
    node_mix<<<N_NODES, 128, 0, stream>>>(meanA, reg_e, ge2b, catb);
    // 37. final: out = lrelu(cat @ spat_w + spat_b)
    wmma_gemm_f32<<<625, 32, 0, stream>>>(catb, 384, spat_w, spat_b, out, 128, 384, 2);
}